// GCN_17815524343826
// MI455X (gfx1250) — compile-verified
//
#include <hip/hip_runtime.h>
#include <hip/hip_bf16.h>

typedef __attribute__((ext_vector_type(2))) float v2f;
typedef __attribute__((ext_vector_type(8))) float v8f;

// ---------------------------------------------------------------------------
// small utility kernels
// ---------------------------------------------------------------------------
__global__ void k_fill(float* __restrict__ p, float v, long long n) {
    long long i = (long long)blockIdx.x * blockDim.x + threadIdx.x;
    if (i < n) p[i] = v;
}

__global__ void k_count_deg(const long long* __restrict__ dst,
                            float* __restrict__ deg, long long E) {
    long long i = (long long)blockIdx.x * blockDim.x + threadIdx.x;
    if (i < E) atomicAdd(&deg[dst[i]], 1.0f);
}

__global__ void k_rsqrt_inplace(float* __restrict__ p, long long n) {
    long long i = (long long)blockIdx.x * blockDim.x + threadIdx.x;
    if (i < n) p[i] = rsqrtf(p[i]);
}

// ---------------------------------------------------------------------------
// WMMA GEMM: H[nrows, NF] = A[nrows, K] @ W[K, NF]   (fp32, V_WMMA_F32_16X16X4_F32)
// One wave computes a 16 x NF tile; 8 waves (256 thr) per block.
// W is staged TRANSPOSED in LDS: lwT[col][k] with a 2-float pad per column so a
// B fragment (k, k+1 for a fixed col) is one aligned ds_load_b64 into an even
// VGPR pair (no v_mov packing), and the 520-byte column stride spreads the 16
// lanes of each half-wave across distinct LDS banks.
// A-frag (16x4 f32): lanes 0-15 = rows, v0/v1 = K,K+1; lanes 16-31 = K+2,K+3.
// B-frag (4x16 f32): lanes 0-15 = cols, v0/v1 = K,K+1; lanes 16-31 = K+2,K+3.
// C/D (16x16 f32): VGPR r -> row r (lanes 0-15) / row r+8 (lanes 16-31).
// ---------------------------------------------------------------------------
template <int K, int NF>
__global__ void k_gemm_wmma(const float* __restrict__ A,
                            const float* __restrict__ W,
                            float* __restrict__ H, int nrows) {
    constexpr int NT = NF / 16;            // N-tiles of 16
    constexpr int KP = K + 2;              // padded column stride (floats)
    __shared__ float lwT[NF * KP];
    for (int i = threadIdx.x; i < K * NF; i += blockDim.x) {
        const int k = i / NF;
        const int n = i % NF;
        lwT[n * KP + k] = W[i];            // transpose into LDS
    }
    __syncthreads();

    const int lane = threadIdx.x & 31;
    const int wave = threadIdx.x >> 5;
    const int tile = blockIdx.x * 8 + wave;
    const int tiles = (nrows + 15) >> 4;
    if (tile >= tiles) return;

    const int mrow  = tile * 16 + (lane & 15);
    const int rowl  = mrow < nrows ? mrow : (nrows - 1);   // clamp for tail tile
    const int khalf = (lane >> 4) * 2;                     // 0 or 2
    const float* __restrict__ arow = A + (long long)rowl * K;

    v8f acc[NT];
#pragma unroll
    for (int t = 0; t < NT; ++t)
#pragma unroll
        for (int r = 0; r < 8; ++r) acc[t][r] = 0.0f;

#pragma unroll 4
    for (int ks = 0; ks < K / 4; ++ks) {
        const int kc = ks * 4 + khalf;                     // even -> 8B aligned
        const v2f a = *reinterpret_cast<const v2f*>(arow + kc);
#pragma unroll
        for (int t = 0; t < NT; ++t) {
            const int col = t * 16 + (lane & 15);
            const v2f b = *reinterpret_cast<const v2f*>(&lwT[col * KP + kc]);
            acc[t] = __builtin_amdgcn_wmma_f32_16x16x4_f32(
                false, a, false, b, (short)0, acc[t], false, false);
        }
    }

    const int rbase = tile * 16 + (lane >> 4) * 8;
#pragma unroll
    for (int t = 0; t < NT; ++t) {
        const int col = t * 16 + (lane & 15);
#pragma unroll
        for (int r = 0; r < 8; ++r) {
            const int row = rbase + r;
            if (row < nrows) H[(long long)row * NF + col] = acc[t][r];
        }
    }
}

// ---------------------------------------------------------------------------
// Edge scatter: agg[dst] += h[src] * dinv[src]*dinv[dst]
// NF/4 lanes per edge, float4 gather + 4 fp32 atomics (L2-resident targets).
// ---------------------------------------------------------------------------
template <int NF>
__global__ void k_scatter(const long long* __restrict__ src,
                          const long long* __restrict__ dst,
                          const float* __restrict__ dinv,
                          const float* __restrict__ h,
                          float* __restrict__ agg, long long E) {
    constexpr int TPE = NF / 4;
    const long long gid = (long long)blockIdx.x * blockDim.x + threadIdx.x;
    const long long e = gid / TPE;
    if (e >= E) return;
    const int f = (int)(gid % TPE) * 4;
    const long long s = src[e];
    const long long d = dst[e];
    const float nrm = dinv[s] * dinv[d];
    const float4 hv = *reinterpret_cast<const float4*>(h + s * NF + f);
    float* ap = agg + d * NF + f;
    atomicAdd(ap + 0, hv.x * nrm);
    atomicAdd(ap + 1, hv.y * nrm);
    atomicAdd(ap + 2, hv.z * nrm);
    atomicAdd(ap + 3, hv.w * nrm);
}

// agg + h*dinv^2 + b, ReLU, write back into h (layer-1 epilogue)
__global__ void k_post1(const float* __restrict__ agg, float* __restrict__ h,
                        const float* __restrict__ dinv,
                        const float* __restrict__ b, long long n) {
    const long long idx = (long long)blockIdx.x * blockDim.x + threadIdx.x;
    if (idx >= n) return;
    const long long i = idx >> 6;     // /64
    const int f = (int)(idx & 63);
    const float di = dinv[i];
    const float v = agg[idx] + h[idx] * di * di + b[f];
    h[idx] = fmaxf(v, 0.0f);
}

// layer-2 epilogue + log_softmax over 32 features; one wave32 per node
__global__ void k_final(const float* __restrict__ agg,
                        const float* __restrict__ h,
                        const float* __restrict__ dinv,
                        const float* __restrict__ b,
                        float* __restrict__ out, int N) {
    const int lane = threadIdx.x & 31;
    const int node = blockIdx.x * 8 + (threadIdx.x >> 5);
    if (node >= N) return;
    const float di = dinv[node];
    const long long idx = (long long)node * 32 + lane;
    const float v = agg[idx] + h[idx] * di * di + b[lane];

    float m = v;
#pragma unroll
    for (int off = 16; off >= 1; off >>= 1)
        m = fmaxf(m, __shfl_xor(m, off, 32));
    float s = expf(v - m);
#pragma unroll
    for (int off = 16; off >= 1; off >>= 1)
        s += __shfl_xor(s, off, 32);
    out[idx] = (v - m) - logf(s);
}

// ---------------------------------------------------------------------------
// launcher
// ---------------------------------------------------------------------------
extern "C" void kernel_launch(void* const* d_in, const int* in_sizes, int n_in,
                              void* d_out, int out_size, void* d_ws, size_t ws_size,
                              hipStream_t stream) {
    const float*     x   = (const float*)d_in[0];
    const long long* ei  = (const long long*)d_in[1];   // int64 [2, E]
    const float*     W1  = (const float*)d_in[2];
    const float*     b1  = (const float*)d_in[3];
    const float*     W2  = (const float*)d_in[4];
    const float*     b2  = (const float*)d_in[5];
    float*           out = (float*)d_out;

    const long long N = in_sizes[0] / 128;
    const long long E = in_sizes[1] / 2;
    const long long* srcI = ei;
    const long long* dstI = ei + E;

    auto al4 = [](long long v) { return (v + 3) & ~3LL; };
    float* ws = (float*)d_ws;
    const long long off_dinv = 0;
    const long long off_h1   = off_dinv + al4(N);
    const long long off_agg1 = off_h1   + al4(N * 64);
    const long long off_agg2 = off_agg1 + al4(N * 64);   // agg1,agg2 contiguous for one zero pass
    const long long off_h2   = off_agg2 + al4(N * 32);
    float* dinv = ws + off_dinv;
    float* h1   = ws + off_h1;
    float* agg1 = ws + off_agg1;
    float* agg2 = ws + off_agg2;
    float* h2   = ws + off_h2;

    const int T = 256;
    auto gs = [&](long long n) { return (unsigned)((n + T - 1) / T); };

    // degrees: deg = 1 + count(dst); dinv = rsqrt(deg)
    k_fill<<<gs(N), T, 0, stream>>>(dinv, 1.0f, N);
    k_count_deg<<<gs(E), T, 0, stream>>>(dstI, dinv, E);
    k_rsqrt_inplace<<<gs(N), T, 0, stream>>>(dinv, N);

    // zero agg1 + agg2 (contiguous region)
    const long long zlen = off_h2 - off_agg1;
    k_fill<<<gs(zlen), T, 0, stream>>>(agg1, 0.0f, zlen);

    // layer 1: h1 = x @ W1 (fp32 WMMA), scatter, epilogue+ReLU
    {
        const int tiles = (int)((N + 15) >> 4);
        const int blocks = (tiles + 7) >> 3;
        k_gemm_wmma<128, 64><<<blocks, T, 0, stream>>>(x, W1, h1, (int)N);
    }
    k_scatter<64><<<gs(E * 16), T, 0, stream>>>(srcI, dstI, dinv, h1, agg1, E);
    k_post1<<<gs(N * 64), T, 0, stream>>>(agg1, h1, dinv, b1, N * 64);

    // layer 2: h2 = relu(h1) @ W2, scatter, epilogue + log_softmax
    {
        const int tiles = (int)((N + 15) >> 4);
        const int blocks = (tiles + 7) >> 3;
        k_gemm_wmma<64, 32><<<blocks, T, 0, stream>>>(h1, W2, h2, (int)N);
    }
    k_scatter<32><<<gs(E * 8), T, 0, stream>>>(srcI, dstI, dinv, h2, agg2, E);
    k_final<<<(unsigned)((N + 7) / 8), T, 0, stream>>>(agg2, h2, dinv, b2, out, (int)N);
}